// SpotlightLoss_85598698209828
// MI455X (gfx1250) — compile-verified
//
#include <hip/hip_runtime.h>

// ---------------- problem constants ----------------
#define RCH   8              // B*C = 4*2 channels
#define SPC   4194304        // 16*512*512 elements per channel
#define NB    256            // histogram bins
#define BPC   64             // blocks/channel for minmax & hist passes
#define BPC4  256            // blocks/channel for main pass
#define EPSF  1e-6f
#define KSIG  (-0.95f)

// ---------------- workspace layout (32-bit words) ----------------
// [0,8)            min enc (uint, order-monotone float encoding)
// [8,16)           max enc (uint)
// [16,2064)        hist   (uint)  8*256
// [2064,2072)      thr    (float)
// [2080,2080+48*256) partials (float) rows q=ch*5+s (q<40); rows 40..47 zero pad
#define W_MIN  0
#define W_MAX  8
#define W_HIST 16
#define W_THR  2064
#define W_PART 2080

typedef __attribute__((ext_vector_type(2))) float v2f;
typedef __attribute__((ext_vector_type(4))) float v4f;
typedef __attribute__((ext_vector_type(8))) float v8f;

// order-preserving float <-> uint (for integer atomicMin/Max => deterministic)
__device__ __forceinline__ unsigned encf(float f) {
  unsigned u = __float_as_uint(f);
  return (u & 0x80000000u) ? ~u : (u | 0x80000000u);
}
__device__ __forceinline__ float decf(unsigned e) {
  unsigned u = (e & 0x80000000u) ? (e & 0x7FFFFFFFu) : ~e;
  return __uint_as_float(u);
}

// ---------------- pass 0: init workspace every launch ----------------
__global__ void k_init(unsigned* __restrict__ ws) {
  int tid = threadIdx.x;
  if (tid < RCH) { ws[W_MIN + tid] = 0xFFFFFFFFu; ws[W_MAX + tid] = 0u; }
  for (int i = tid; i < RCH * NB; i += blockDim.x) ws[W_HIST + i] = 0u;
  float* f = (float*)ws;
  for (int i = tid; i < 8 * NB; i += blockDim.x) f[W_PART + 40 * NB + i] = 0.0f;  // zero pad rows
}

// ---------------- pass 1: per-channel min/max of target ----------------
__global__ void __launch_bounds__(256) k_minmax(const float* __restrict__ tgt,
                                                unsigned* __restrict__ ws) {
  const int ch = blockIdx.y, b = blockIdx.x, tid = threadIdx.x;
  const v4f* t4 = (const v4f*)tgt;
  const long base = (long)ch * (SPC / 4) + (long)b * (SPC / 4 / BPC);  // 16384 v4f / block
  float mn = 3.4e38f, mx = -3.4e38f;
  for (int i = 0; i < 64; ++i) {
    long off = base + (long)i * 256 + tid;
    if (i + 1 < 64) __builtin_prefetch((const void*)(t4 + off + 256), 0, 0);  // global_prefetch
    v4f v = t4[off];
    mn = fminf(mn, fminf(fminf(v.x, v.y), fminf(v.z, v.w)));
    mx = fmaxf(mx, fmaxf(fmaxf(v.x, v.y), fmaxf(v.z, v.w)));
  }
  __shared__ float smn[256], smx[256];
  smn[tid] = mn; smx[tid] = mx; __syncthreads();
  for (int s = 128; s > 0; s >>= 1) {
    if (tid < s) { smn[tid] = fminf(smn[tid], smn[tid + s]); smx[tid] = fmaxf(smx[tid], smx[tid + s]); }
    __syncthreads();
  }
  if (tid == 0) {
    atomicMin(&ws[W_MIN + ch], encf(smn[0]));
    atomicMax(&ws[W_MAX + ch], encf(smx[0]));
  }
}

// ---------------- pass 2: per-channel 256-bin histogram ----------------
__global__ void __launch_bounds__(256) k_hist(const float* __restrict__ tgt,
                                              unsigned* __restrict__ ws) {
  __shared__ unsigned lh[NB];
  const int ch = blockIdx.y, b = blockIdx.x, tid = threadIdx.x;
  for (int i = tid; i < NB; i += 256) lh[i] = 0u;
  __syncthreads();
  const float lo = decf(ws[W_MIN + ch]);
  const float hi = decf(ws[W_MAX + ch]);
  const float span = hi - lo;
  const float ss = (span > 0.0f) ? span : 1.0f;
  const v4f* t4 = (const v4f*)tgt;
  const long base = (long)ch * (SPC / 4) + (long)b * (SPC / 4 / BPC);
  for (int i = 0; i < 64; ++i) {
    v4f v = t4[base + (long)i * 256 + tid];
#pragma unroll
    for (int c = 0; c < 4; ++c) {
      float x = v[c];
      int idx = (int)((x - lo) / ss * 256.0f);            // matches astype(int32) trunc
      idx = min(max(idx, 0), NB - 1);
      atomicAdd(&lh[idx], 1u);
    }
  }
  __syncthreads();
  for (int i = tid; i < NB; i += 256)
    if (lh[i]) atomicAdd(&ws[W_HIST + ch * NB + i], lh[i]);
}

// ---------------- pass 3: Otsu threshold (tiny) ----------------
__global__ void k_otsu(unsigned* __restrict__ ws) {
  const int ch = threadIdx.x;
  if (ch >= RCH) return;
  const float lo = decf(ws[W_MIN + ch]);
  const float hi = decf(ws[W_MAX + ch]);
  const float span = hi - lo;
  const unsigned* h = &ws[W_HIST + ch * NB];
  float total = 0.0f, gmean = 0.0f;
  for (int i = 0; i < NB; ++i) {
    float e0 = lo + span * ((float)i * (1.0f / 256.0f));
    float e1 = lo + span * ((float)(i + 1) * (1.0f / 256.0f));
    float c = (e0 + e1) * 0.5f;
    float hv = (float)h[i];
    total += hv; gmean += hv * c;
  }
  float cs = 0.0f, cm = 0.0f, best = -1.0f; int bi = 0;
  for (int i = 0; i < NB; ++i) {
    float e0 = lo + span * ((float)i * (1.0f / 256.0f));
    float e1 = lo + span * ((float)(i + 1) * (1.0f / 256.0f));
    float c = (e0 + e1) * 0.5f;
    float hv = (float)h[i];
    cs += hv; cm += hv * c;
    float num = cm * total - gmean * cs;
    float iv = num * num / (cs * (total - cs) + 1e-10f);
    if (iv > best) { best = iv; bi = i; }                 // first-max like jnp.argmax
  }
  float e0 = lo + span * ((float)bi * (1.0f / 256.0f));
  float e1 = lo + span * ((float)(bi + 1) * (1.0f / 256.0f));
  float thr = (e0 + e1) * 0.5f;
  if (!(span > 0.0f)) thr = lo;
  ((float*)ws)[W_THR + ch] = thr;
}

// ---------------- pass 4: fused masked-MSE + soft-dice partial sums ----------------
__global__ void __launch_bounds__(256) k_main(const float* __restrict__ pred,
                                              const float* __restrict__ tgt,
                                              const unsigned* __restrict__ ws,
                                              float* __restrict__ part) {
  const int ch = blockIdx.y, b = blockIdx.x, tid = threadIdx.x;
  const float thr = ((const float*)ws)[W_THR + ch];
  const v4f* p4 = (const v4f*)pred;
  const v4f* t4 = (const v4f*)tgt;
  const long base = (long)ch * (SPC / 4) + (long)b * (SPC / 4 / BPC4);  // 4096 v4f / block
  float cnt = 0.0f, msum = 0.0f, tsum = 0.0f, inter = 0.0f, ssum = 0.0f;
  for (int i = 0; i < 16; ++i) {
    long off = base + (long)i * 256 + tid;
    v4f tv = t4[off];                                   // RT: stays in 192MB L2 for reuse
    v4f pv = __builtin_nontemporal_load(p4 + off);      // single-use stream: NT, spare L2
#pragma unroll
    for (int c = 0; c < 4; ++c) {
      float p = pv[c], t = tv[c];
      float d = p - t, sq = d * d;
      tsum += sq;
      float m = (t >= thr) ? 1.0f : 0.0f;
      cnt += m; msum += m * sq;
      float ax = fabsf(p);
      float raw = (p - KSIG * p) / (KSIG - 2.0f * KSIG * ax + 1.0f);
      float sp = fminf(fmaxf(raw, 0.0f), 1.0f);
      ssum += sp; inter += m * sp;
    }
  }
  __shared__ float sd[256];
  float vals[5] = {cnt, msum, tsum, inter, ssum};
#pragma unroll
  for (int s5 = 0; s5 < 5; ++s5) {
    sd[tid] = vals[s5]; __syncthreads();
    for (int st = 128; st > 0; st >>= 1) {
      if (tid < st) sd[tid] += sd[tid + st];
      __syncthreads();
    }
    if (tid == 0) part[(ch * 5 + s5) * NB + b] = sd[0];
    __syncthreads();
  }
}

// ---------------- pass 5: WMMA f32 reduction P(48x256) * ones(256) + combine ----------------
__global__ void k_final(const float* __restrict__ part, float* __restrict__ out) {
  const int lane = threadIdx.x;          // 32 threads = 1 wave, EXEC all ones
  __shared__ float s48[48];
  const int m  = lane & 15;              // A-matrix row (M)
  const int kb = (lane >> 4) << 1;       // lanes 0-15 -> K0/K1, lanes 16-31 -> K2/K3
  v2f bones; bones.x = 1.0f; bones.y = 1.0f;   // B = ones(4x16)
  for (int tile = 0; tile < 3; ++tile) {
    v8f c = {0.f, 0.f, 0.f, 0.f, 0.f, 0.f, 0.f, 0.f};
    const float* row = part + (tile * 16 + m) * NB + kb;   // rows 40..47 are zero pad
    for (int k0 = 0; k0 < NB; k0 += 4) {
      v2f a; a.x = row[k0]; a.y = row[k0 + 1];
      // D[m][n] = sum_k A[m][k]*1 + C : accumulates 256-wide row sums, f32-exact
      c = __builtin_amdgcn_wmma_f32_16x16x4_f32(false, a, false, bones,
                                                (short)0, c, false, false);
    }
    if (lane == 0 || lane == 16) {       // column N=0: VGPR v -> M=v (lane0) / M=8+v (lane16)
      int m0 = (lane >> 4) * 8;
#pragma unroll
      for (int v = 0; v < 8; ++v) s48[tile * 16 + m0 + v] = c[v];
    }
    __syncthreads();
  }
  if (lane == 0) {
    float mse_acc = 0.0f, dice_acc = 0.0f; int nmask = 0;
    for (int ch = 0; ch < RCH; ++ch) {
      float cnt   = s48[ch * 5 + 0];
      float msum  = s48[ch * 5 + 1];
      float tsum  = s48[ch * 5 + 2];
      float inter = s48[ch * 5 + 3];
      float ssum  = s48[ch * 5 + 4];
      float rmse = tsum * (1.0f / (float)SPC);
      bool hf = cnt > 0.0f;
      float cmse = hf ? (msum / (cnt + EPSF)) : rmse;
      mse_acc += cmse;
      float cdice = 1.0f - 2.0f * inter / (ssum + cnt + EPSF);
      if (hf) { dice_acc += cdice; ++nmask; }
    }
    float mmse = mse_acc * (1.0f / (float)RCH);
    float dice = (nmask > 0) ? dice_acc / (float)nmask : 0.0f;
    out[0] = 0.5f * mmse + 0.5f * dice;
  }
}

extern "C" void kernel_launch(void* const* d_in, const int* in_sizes, int n_in,
                              void* d_out, int out_size, void* d_ws, size_t ws_size,
                              hipStream_t stream) {
  const float* pred = (const float*)d_in[0];
  const float* tgt  = (const float*)d_in[1];
  float* out  = (float*)d_out;
  unsigned* ws = (unsigned*)d_ws;
  float* part = (float*)d_ws + W_PART;
  (void)in_sizes; (void)n_in; (void)out_size; (void)ws_size;

  k_init  <<<1, 256, 0, stream>>>(ws);
  k_minmax<<<dim3(BPC,  RCH), 256, 0, stream>>>(tgt, ws);
  k_hist  <<<dim3(BPC,  RCH), 256, 0, stream>>>(tgt, ws);
  k_otsu  <<<1, 32, 0, stream>>>(ws);
  k_main  <<<dim3(BPC4, RCH), 256, 0, stream>>>(pred, tgt, ws, part);
  k_final <<<1, 32, 0, stream>>>(part, out);
}